// Metric_81595788690028
// MI455X (gfx1250) — compile-verified
//
#include <hip/hip_runtime.h>

// B=16, C=3, H=768, W=1024 — fixed by the reference setup.
#define Bn 16
#define Hn 768
#define Wn 1024
#define HW (Hn * Wn)

typedef float f4 __attribute__((ext_vector_type(4)));

__global__ __launch_bounds__(256) void warp_l1_mean_kernel(
    const float* __restrict__ first,   // [B,3,H,W]
    const float* __restrict__ second,  // [B,3,H,W]
    const float* __restrict__ flow,    // [B,2,H,W]
    const float* __restrict__ alpha_p, // [1]
    float* __restrict__ out)           // [B,1,H,W]
{
    // 4 pixels per lane along W (W % 4 == 0, so a float4 never crosses a row)
    const unsigned v   = blockIdx.x * blockDim.x + threadIdx.x;
    const unsigned p   = v * 4u;                    // base pixel index into [B*H*W]
    const unsigned b   = p / (unsigned)HW;
    const unsigned rem = p - b * (unsigned)HW;      // h*W + w
    const unsigned h   = rem / (unsigned)Wn;
    const unsigned w   = rem - h * (unsigned)Wn;

    const float alpha = alpha_p[0];

    // Streaming inputs: non-temporal so the 151MB gather target (ten_second)
    // stays resident in the 192MB L2.
    const float* fxp = flow + (size_t)b * 2u * HW + rem;
    const f4 fx = __builtin_nontemporal_load((const f4*)fxp);
    const f4 fy = __builtin_nontemporal_load((const f4*)(fxp + HW));

    const float* f0p = first + (size_t)b * 3u * HW + rem;
    const f4 a0 = __builtin_nontemporal_load((const f4*)(f0p));
    const f4 a1 = __builtin_nontemporal_load((const f4*)(f0p + HW));
    const f4 a2 = __builtin_nontemporal_load((const f4*)(f0p + 2 * HW));

    const float* sb = second + (size_t)b * 3u * HW;

    const float SX = (float)Wn / (float)(Wn - 1);   // flow -> pixel scale, exact algebra
    const float SY = (float)Hn / (float)(Hn - 1);
    const float hf = (float)h;

    f4 res;
#pragma unroll
    for (int i = 0; i < 4; ++i) {
        const float ix = (float)(w + (unsigned)i) + fx[i] * SX;
        const float iy = hf + fy[i] * SY;

        const float x0f = floorf(ix), y0f = floorf(iy);
        const float wx1 = ix - x0f,   wy1 = iy - y0f;
        const float wx0 = 1.0f - wx1, wy0 = 1.0f - wy1;
        const float x1f = x0f + 1.0f, y1f = y0f + 1.0f;

        // zero-padding validity per tap (matches grid_sample padding_mode='zeros')
        const bool vx0 = (x0f >= 0.0f) && (x0f <= (float)(Wn - 1));
        const bool vx1 = (x1f >= 0.0f) && (x1f <= (float)(Wn - 1));
        const bool vy0 = (y0f >= 0.0f) && (y0f <= (float)(Hn - 1));
        const bool vy1 = (y1f >= 0.0f) && (y1f <= (float)(Hn - 1));

        const float w00 = (vx0 && vy0) ? (wy0 * wx0) : 0.0f;
        const float w10 = (vx1 && vy0) ? (wy0 * wx1) : 0.0f;
        const float w01 = (vx0 && vy1) ? (wy1 * wx0) : 0.0f;
        const float w11 = (vx1 && vy1) ? (wy1 * wx1) : 0.0f;

        // clamp (safe addresses; invalid taps already have weight 0)
        const int x0 = (int)fminf(fmaxf(x0f, 0.0f), (float)(Wn - 1));
        const int x1 = (int)fminf(fmaxf(x1f, 0.0f), (float)(Wn - 1));
        const int y0 = (int)fminf(fmaxf(y0f, 0.0f), (float)(Hn - 1));
        const int y1 = (int)fminf(fmaxf(y1f, 0.0f), (float)(Hn - 1));

        const int i00 = y0 * Wn + x0;
        const int i10 = y0 * Wn + x1;
        const int i01 = y1 * Wn + x0;
        const int i11 = y1 * Wn + x1;

        const float af[3] = { a0[i], a1[i], a2[i] };
        float acc = 0.0f;
#pragma unroll
        for (int c = 0; c < 3; ++c) {
            const float* __restrict__ sc = sb + c * HW;   // gathers use regular (RT) caching
            const float wv = sc[i00] * w00 + sc[i10] * w10
                           + sc[i01] * w01 + sc[i11] * w11;
            acc += fabsf(af[c] - wv);
        }
        res[i] = alpha * (acc * (1.0f / 3.0f));
    }

    __builtin_nontemporal_store(res, (f4*)(out + p));
}

extern "C" void kernel_launch(void* const* d_in, const int* in_sizes, int n_in,
                              void* d_out, int out_size, void* d_ws, size_t ws_size,
                              hipStream_t stream) {
    (void)in_sizes; (void)n_in; (void)out_size; (void)d_ws; (void)ws_size;
    const float* first  = (const float*)d_in[0];
    const float* second = (const float*)d_in[1];
    const float* flow   = (const float*)d_in[2];
    const float* alpha  = (const float*)d_in[3];
    float* out = (float*)d_out;

    const unsigned total_vec = (unsigned)Bn * HW / 4u;   // 3,145,728 lanes
    const unsigned block = 256;                           // 8 wave32s
    const unsigned grid  = total_vec / block;             // 12,288 blocks (exact)
    warp_l1_mean_kernel<<<grid, block, 0, stream>>>(first, second, flow, alpha, out);
}